// StabilityLoss_11665131176104
// MI455X (gfx1250) — compile-verified
//
#include <hip/hip_runtime.h>

typedef float v2f __attribute__((ext_vector_type(2)));
typedef float v8f __attribute__((ext_vector_type(8)));

#define N_ROWS 8192
#define D_COLS 4096
#define CHUNKS 64
#define ROWS_PER_CHUNK (N_ROWS / CHUNKS)       // 128
#define K1_BLOCK 256
#define K1_GRID_X (D_COLS / (K1_BLOCK * 4))    // 4
#define NPSQ (K1_GRID_X * CHUNKS)              // 256

// ws float layout:
//   [0, CHUNKS*D)            partial column sums  (1 MB)
//   [CHUNKS*D, +D)           colsum[4096]
//   [CHUNKS*D + D, +NPSQ)    per-block sum-of-squares partials

// ---------------- Kernel 1: streaming pass (bandwidth-bound, ~5.8us floor) ---
__global__ void k1_partial(const float* __restrict__ X,
                           float* __restrict__ pcol,
                           float* __restrict__ psq) {
    const int tid   = threadIdx.x;
    const int col4  = (blockIdx.x * K1_BLOCK + tid) * 4;  // 4 columns per thread
    const int chunk = blockIdx.y;
    const int r0    = chunk * ROWS_PER_CHUNK;

    const float4* Xv  = reinterpret_cast<const float4*>(X);
    const int rowStep = D_COLS >> 2;
    int base = r0 * rowStep + (col4 >> 2);

    float sx = 0.f, sy = 0.f, sz = 0.f, sw = 0.f;
    float q = 0.f;
#pragma unroll 8
    for (int r = 0; r < ROWS_PER_CHUNK; ++r) {
        float4 v = Xv[base + r * rowStep];          // global_load_b128, coalesced
        sx += v.x; sy += v.y; sz += v.z; sw += v.w;
        q = fmaf(v.x, v.x, q);
        q = fmaf(v.y, v.y, q);
        q = fmaf(v.z, v.z, q);
        q = fmaf(v.w, v.w, q);
    }
    float4 s = make_float4(sx, sy, sz, sw);
    reinterpret_cast<float4*>(pcol)[(chunk * D_COLS + col4) >> 2] = s;

    // deterministic block tree-reduction of sum-of-squares
    __shared__ float red[K1_BLOCK];
    red[tid] = q;
    __syncthreads();
    for (int off = K1_BLOCK / 2; off > 0; off >>= 1) {
        if (tid < off) red[tid] += red[tid + off];
        __syncthreads();
    }
    if (tid == 0) psq[blockIdx.y * gridDim.x + blockIdx.x] = red[0];
}

// ---------------- Kernel 2: fold 64 chunk partials -> colsum[D] --------------
__global__ void k2_colsum(const float* __restrict__ pcol,
                          float* __restrict__ colsum) {
    const int c = blockIdx.x * blockDim.x + threadIdx.x;   // 0..4095
    float s = 0.f;
#pragma unroll 8
    for (int k = 0; k < CHUNKS; ++k) s += pcol[k * D_COLS + c];
    colsum[c] = s;
}

// ---------------- Kernel 3: one wave32; WMMA self-dot of colsum --------------
// dot(colsum,colsum) via V_WMMA_F32_16X16X4_F32: per-iteration, row m of A and
// col m of B both hold colsum[base+4m .. base+4m+3], so A and B share the same
// 2-VGPR register image; the D diagonal accumulates sum_k colsum[base+4m+k]^2.
__global__ void k3_final(const float* __restrict__ colsum,
                         const float* __restrict__ psq,
                         float* __restrict__ out) {
    const int lane = threadIdx.x;          // exactly 32 threads -> EXEC all 1s

    // total sum of squares (uniform trip count, no divergence)
    float q = 0.f;
#pragma unroll
    for (int i = 0; i < NPSQ / 32; ++i) q += psq[i * 32 + lane];

    // WMMA accumulate: 64 iterations cover all 4096 colsum entries
    const int lp = lane & 15;
    const int hi = (lane >> 4) & 1;        // 0: K=0,1 ; 1: K=2,3
    v8f acc = {};
    for (int base = 0; base < D_COLS; base += 64) {
        const int idx = base + 4 * lp + 2 * hi;
        v2f a;
        a.x = colsum[idx];
        a.y = colsum[idx + 1];
        acc = __builtin_amdgcn_wmma_f32_16x16x4_f32(
            /*neg_a=*/false, a, /*neg_b=*/false, a,
            /*c_mod=*/(short)0, acc, /*reuse_a=*/false, /*reuse_b=*/false);
    }

    // wave-reduce sum of squares
    for (int off = 16; off > 0; off >>= 1) q += __shfl_xor(q, off, 32);

    // extract diagonal of D (C/D layout: VGPR r, lanes0-15 -> M=r,N=lane;
    // lanes16-31 -> M=r+8,N=lane-16). Diag entries live at:
    //   m in [0,8):  VGPR m,      lane m
    //   m in [8,16): VGPR m-8,    lane m+16  (i.e. lanes 24..31, VGPR lane-24)
    float arr[8];
#pragma unroll
    for (int i = 0; i < 8; ++i) arr[i] = acc[i];
    float diag = 0.f;
    if (lane < 8)        diag = arr[lane];
    else if (lane >= 24) diag = arr[lane - 24];
    for (int off = 16; off > 0; off >>= 1) diag += __shfl_xor(diag, off, 32);

    if (lane == 0) out[0] = (float)N_ROWS * q - diag;
}

extern "C" void kernel_launch(void* const* d_in, const int* in_sizes, int n_in,
                              void* d_out, int out_size, void* d_ws, size_t ws_size,
                              hipStream_t stream) {
    (void)in_sizes; (void)n_in; (void)out_size; (void)ws_size;
    const float* X = (const float*)d_in[0];
    float* ws     = (float*)d_ws;
    float* pcol   = ws;                         // CHUNKS*D floats
    float* colsum = ws + CHUNKS * D_COLS;       // D floats
    float* psq    = colsum + D_COLS;            // NPSQ floats
    float* out    = (float*)d_out;

    dim3 g1(K1_GRID_X, CHUNKS);
    k1_partial<<<g1, K1_BLOCK, 0, stream>>>(X, pcol, psq);
    k2_colsum<<<D_COLS / 256, 256, 0, stream>>>(pcol, colsum);
    k3_final<<<1, 32, 0, stream>>>(colsum, psq, out);
}